// KirchhoffNormalisation_83786222010480
// MI455X (gfx1250) — compile-verified
//
#include <hip/hip_runtime.h>
#include <hip/hip_bf16.h>
#include <math.h>

#define BATCH 16
#define NDIM  512
#define NT    (NDIM / 16)   // 32 tiles per dimension
#define HDIM  8
#define NEG_INF (-87.336544750402326f)
#define SMOOTH  0.0005f

typedef float v2f __attribute__((ext_vector_type(2)));
typedef float v8f __attribute__((ext_vector_type(8)));
typedef unsigned int u32x4 __attribute__((ext_vector_type(4)));
typedef int i32x8 __attribute__((ext_vector_type(8)));
typedef int i32x4 __attribute__((ext_vector_type(4)));

#if defined(__has_builtin)
#if __has_builtin(__builtin_amdgcn_tensor_load_to_lds) && \
    __has_builtin(__builtin_amdgcn_s_wait_tensorcnt)
#define USE_TDM 1
#endif
#endif

// ---------------------------------------------------------------------------
// Kernel 1: dep_score = logsumexp_h(masked hds); build Laplacian L.
// ---------------------------------------------------------------------------
__global__ __launch_bounds__(256)
void build_l_kernel(const float* __restrict__ hds,
                    const unsigned char* __restrict__ mask,
                    float* __restrict__ dep,    // [B,N,N]
                    float* __restrict__ Lmat)   // [B,N,N]
{
  const int b = blockIdx.x / NDIM;
  const int i = blockIdx.x % NDIM;
  const int tid = threadIdx.x;
  __shared__ float red[256];

  const bool mi = mask[b * NDIM + i] != 0;

  float Aval[2], dsv[2];
  float rowsum = 0.f;
  #pragma unroll
  for (int u = 0; u < 2; ++u) {
    const int j = tid + u * 256;
    const bool mj = mask[b * NDIM + j] != 0;
    const bool dm = (i == j) || (!mi) || (!mj);
    const float* p = hds + (((size_t)b * NDIM + i) * NDIM + j) * HDIM;
    float v[HDIM];
    float mx = NEG_INF;
    #pragma unroll
    for (int h = 0; h < HDIM; ++h) {
      float x = dm ? NEG_INF : p[h];
      v[h] = x;
      mx = fmaxf(mx, x);
    }
    float s = 0.f;
    #pragma unroll
    for (int h = 0; h < HDIM; ++h) s += __expf(v[h] - mx);
    const float ds = mx + __logf(s);
    dsv[u] = ds;
    const float A = dm ? 0.f : (__expf(ds) + SMOOTH);
    Aval[u] = A;
    rowsum += A;
  }

  red[tid] = rowsum;
  __syncthreads();
  for (int st = 128; st > 0; st >>= 1) {
    if (tid < st) red[tid] += red[tid + st];
    __syncthreads();
  }
  rowsum = red[0];

  const float rho = mi ? (float)HDIM : 0.f;

  #pragma unroll
  for (int u = 0; u < 2; ++u) {
    const int j = tid + u * 256;
    const bool mj = mask[b * NDIM + j] != 0;
    float Lv;
    if (j == 0)      Lv = rho;
    else if (j == i) Lv = rowsum;
    else             Lv = -Aval[u];
    if ((!mi) || (!mj))     Lv = 0.f;
    if ((i == j) && (!mj))  Lv = 1.f;
    const size_t idx = ((size_t)b * NDIM + i) * NDIM + j;
    dep[idx]  = dsv[u];
    Lmat[idx] = Lv;
  }
}

// ---------------------------------------------------------------------------
// Kernel 2a (panel, per GJ step kb): invert 16x16 diag block in LDS,
// accumulate logdet, compute new pivot row block Rnew = D^-1 * Rold.
// Row block staged to LDS via the Tensor Data Mover, overlapped with the
// diagonal-block inversion; completion via s_wait_tensorcnt.
// ---------------------------------------------------------------------------
__global__ __launch_bounds__(256)
void panel_kernel(float* __restrict__ Mbase, float* __restrict__ logdetw, int kb)
{
  const int b = blockIdx.x;
  float* __restrict__ M = Mbase + (size_t)b * NDIM * NDIM;
  const int tid  = threadIdx.x;
  const int K    = kb * 16;

  __shared__ float sD[16 * 16];
  __shared__ float sOld[16 * NDIM];
  __shared__ float sLog;
  if (tid == 0) sLog = 0.f;

  const int r = tid >> 4, c = tid & 15;
  sD[tid] = M[(size_t)(K + r) * NDIM + K + c];

#ifdef USE_TDM
  // Wave 0 kicks off an async TDM copy of the 16x512 pivot row block into
  // LDS; it overlaps with the in-LDS diagonal-block inversion below.
  if (tid < 32) {
    const unsigned int lds_off = (unsigned int)(uintptr_t)(void*)&sOld[0];
    const unsigned long long ga =
        (unsigned long long)(uintptr_t)(M + (size_t)K * NDIM);
    u32x4 g0;
    g0[0] = 1u;                                   // count=1, user descriptor
    g0[1] = lds_off;                              // lds_addr
    g0[2] = (unsigned int)(ga & 0xffffffffull);   // global_addr[31:0]
    g0[3] = (unsigned int)((ga >> 32) & 0x1ffffffull) | (2u << 30); // [56:32] | type=2
    i32x8 g1;
    g1[0] = 0x20000;      // workgroup_mask=0, data_size=2 (4 bytes)
    g1[1] = 0x02000000;   // tensor_dim0 = 512 (bits 79:48)
    g1[2] = 0x02000000;   // tensor_dim1 = 512 (bits 111:80)
    g1[3] = 0x02000000;   // tile_dim0   = 512 (bits 127:112)
    g1[4] = 16;           // tile_dim1   = 16
    g1[5] = 512;          // tensor_dim0_stride = 512
    g1[6] = 0;
    g1[7] = 0;
    i32x4 z4; z4[0] = 0; z4[1] = 0; z4[2] = 0; z4[3] = 0;
    i32x8 z8; z8[0] = 0; z8[1] = 0; z8[2] = 0; z8[3] = 0;
    z8[4] = 0; z8[5] = 0; z8[6] = 0; z8[7] = 0;
    __builtin_amdgcn_tensor_load_to_lds(g0, g1, z4, z4, z8, 0);
  }
#endif
  __syncthreads();

  // Unblocked Gauss-Jordan inversion of sD with logdet accumulation.
  for (int k = 0; k < 16; ++k) {
    const float piv = sD[k * 16 + k];
    const float pin = 1.f / piv;
    float nv;
    if (r == k && c == k) nv = pin;
    else if (r == k)      nv =  sD[k * 16 + c] * pin;
    else if (c == k)      nv = -sD[r * 16 + k] * pin;
    else                  nv =  sD[r * 16 + c] - sD[r * 16 + k] * sD[k * 16 + c] * pin;
    __syncthreads();
    sD[r * 16 + c] = nv;
    if (tid == 0) sLog += __logf(fabsf(piv));
    __syncthreads();
  }

#ifdef USE_TDM
  if (tid < 32) __builtin_amdgcn_s_wait_tensorcnt(0);
#else
  for (int idx = tid; idx < 16 * NDIM; idx += 256) {
    const int rr = idx >> 9, cc = idx & (NDIM - 1);
    sOld[idx] = M[(size_t)(K + rr) * NDIM + cc];
  }
#endif
  __syncthreads();

  // Rnew = D^-1 * Rold; K columns of the row block become D^-1 itself.
  for (int idx = tid; idx < 16 * NDIM; idx += 256) {
    const int rr = idx >> 9, cc = idx & (NDIM - 1);
    float v;
    if ((cc >> 4) == kb) {
      v = sD[rr * 16 + (cc & 15)];
    } else {
      float acc = 0.f;
      #pragma unroll
      for (int t = 0; t < 16; ++t) acc += sD[rr * 16 + t] * sOld[t * NDIM + cc];
      v = acc;
    }
    M[(size_t)(K + rr) * NDIM + cc] = v;
  }

  if (tid == 0) {
    const float prev = (kb == 0) ? 0.f : logdetw[b];
    logdetw[b] = prev + sLog;
  }
}

// ---------------------------------------------------------------------------
// Kernel 2b (trailing, per GJ step kb): A[i][j] -= A[i][K] * Rnew[K][j] for
// all row tiles it != kb; K-column tile becomes -A[i][K]*D^-1 (C seeded 0).
// Grid: BATCH * 31 row-tile blocks; 8 waves x 4 column tiles each.
// WMMA: 4 chained V_WMMA_F32_16X16X4_F32 per 16x16 output tile.
// ---------------------------------------------------------------------------
__global__ __launch_bounds__(256)
void trailing_kernel(float* __restrict__ Mbase, int kb)
{
  const int b   = blockIdx.x / (NT - 1);
  const int itx = blockIdx.x % (NT - 1);
  const int it  = itx + (itx >= kb ? 1 : 0);
  float* __restrict__ M = Mbase + (size_t)b * NDIM * NDIM;

  const int tid  = threadIdx.x;
  const int lane = tid & 31;
  const int wave = tid >> 5;
  const int K    = kb * 16;
  const int rb   = it * 16;

  const int m    = lane & 15;
  const int koff = (lane < 16) ? 0 : 2;
  const int mr   = (lane < 16) ? 0 : 8;

  // Load (negated) A-fragments covering the 16x16 column block BEFORE the
  // barrier; the K-column tile is overwritten after it.
  v2f afr[4];
  #pragma unroll
  for (int t = 0; t < 4; ++t) {
    afr[t][0] = -M[(size_t)(rb + m) * NDIM + K + 4 * t + koff];
    afr[t][1] = -M[(size_t)(rb + m) * NDIM + K + 4 * t + koff + 1];
  }
  __syncthreads();

  #pragma unroll
  for (int q = 0; q < 4; ++q) {
    const int jt = wave * 4 + q;
    const int cb = jt * 16;
    v8f c;
    if (jt == kb) {
      c = (v8f){0.f, 0.f, 0.f, 0.f, 0.f, 0.f, 0.f, 0.f};
    } else {
      #pragma unroll
      for (int rr = 0; rr < 8; ++rr)
        c[rr] = M[(size_t)(rb + mr + rr) * NDIM + cb + m];
    }
    #pragma unroll
    for (int t = 0; t < 4; ++t) {
      v2f bfr;
      bfr[0] = M[(size_t)(K + 4 * t + koff) * NDIM + cb + m];
      bfr[1] = M[(size_t)(K + 4 * t + koff + 1) * NDIM + cb + m];
      c = __builtin_amdgcn_wmma_f32_16x16x4_f32(
              false, afr[t], false, bfr, (short)0, c, false, false);
    }
    #pragma unroll
    for (int rr = 0; rr < 8; ++rr)
      M[(size_t)(rb + mr + rr) * NDIM + cb + m] = c[rr];
  }
}

// ---------------------------------------------------------------------------
// Kernel 3: gradients from the inverse.
// ---------------------------------------------------------------------------
__global__ __launch_bounds__(256)
void grad_kernel(const float* __restrict__ hds,
                 const unsigned char* __restrict__ mask,
                 const float* __restrict__ dep,
                 const float* __restrict__ Minv,
                 float* __restrict__ out_hp,
                 float* __restrict__ rowent)
{
  const int b = blockIdx.x / NDIM;
  const int i = blockIdx.x % NDIM;
  const int tid = threadIdx.x;
  __shared__ float red[256];
  __shared__ float sGd;

  const float* Mb = Minv + (size_t)b * NDIM * NDIM;
  if (tid == 0) sGd = Mb[(size_t)i * NDIM + i];
  __syncthreads();

  const bool  mi    = mask[b * NDIM + i] != 0;
  const float gdiag = (i != 0) ? sGd : 0.f;

  float ent = 0.f;
  #pragma unroll
  for (int u = 0; u < 2; ++u) {
    const int j = tid + u * 256;
    const bool mj = mask[b * NDIM + j] != 0;
    const size_t idx = ((size_t)b * NDIM + i) * NDIM + j;
    const float gij   = (j != 0) ? Mb[(size_t)j * NDIM + i] : 0.f;
    const float dabar = gdiag - gij;
    const float ds    = dep[idx];
    const bool  dm    = (i == j) || (!mi) || (!mj);
    const float dds   = dm ? 0.f : dabar * __expf(ds);
    const float* p = hds + idx * HDIM;
    float* o = out_hp + idx * HDIM;
    if (dm) {
      #pragma unroll
      for (int h = 0; h < HDIM; ++h) o[h] = 0.f;
    } else {
      #pragma unroll
      for (int h = 0; h < HDIM; ++h) o[h] = dds * __expf(p[h] - ds);
    }
    ent += dds * ds;
  }

  red[tid] = ent;
  __syncthreads();
  for (int st = 128; st > 0; st >>= 1) {
    if (tid < st) red[tid] += red[tid + st];
    __syncthreads();
  }
  if (tid == 0) rowent[b * NDIM + i] = red[0];
}

// ---------------------------------------------------------------------------
// Kernel 4: entropy[b] = logdet[b] - sum_i rowent[b,i]; head_root = 0.
// ---------------------------------------------------------------------------
__global__ __launch_bounds__(512)
void finalize_kernel(const float* __restrict__ logdetw,
                     const float* __restrict__ rowent,
                     float* __restrict__ out_hr,
                     float* __restrict__ out_ent)
{
  const int b = blockIdx.x;
  const int tid = threadIdx.x;
  __shared__ float red[512];
  red[tid] = rowent[b * NDIM + tid];
  __syncthreads();
  for (int st = 256; st > 0; st >>= 1) {
    if (tid < st) red[tid] += red[tid + st];
    __syncthreads();
  }
  if (tid == 0) out_ent[b] = logdetw[b] - red[0];
  for (int idx = tid; idx < NDIM * HDIM; idx += 512)
    out_hr[(size_t)b * NDIM * HDIM + idx] = 0.f;
}

// ---------------------------------------------------------------------------
extern "C" void kernel_launch(void* const* d_in, const int* in_sizes, int n_in,
                              void* d_out, int out_size, void* d_ws, size_t ws_size,
                              hipStream_t stream)
{
  const float* hds = (const float*)d_in[0];
  const unsigned char* mask = (const unsigned char*)d_in[2];
  (void)d_in[1]; (void)in_sizes; (void)n_in; (void)out_size; (void)ws_size;

  const size_t BNN = (size_t)BATCH * NDIM * NDIM;
  float* w       = (float*)d_ws;
  float* dep     = w;                 // B*N*N
  float* Mmat    = w + BNN;           // B*N*N (Laplacian -> inverse, in place)
  float* logdetw = w + 2 * BNN;       // B
  float* rowent  = logdetw + BATCH;   // B*N

  float* out_hp  = (float*)d_out;                        // [B,N,N,H]
  float* out_hr  = out_hp + BNN * HDIM;                  // [B,N,H]
  float* out_ent = out_hr + (size_t)BATCH * NDIM * HDIM; // [B]

  build_l_kernel<<<BATCH * NDIM, 256, 0, stream>>>(hds, mask, dep, Mmat);

  for (int kb = 0; kb < NT; ++kb) {
    panel_kernel   <<<BATCH,            256, 0, stream>>>(Mmat, logdetw, kb);
    trailing_kernel<<<BATCH * (NT - 1), 256, 0, stream>>>(Mmat, kb);
  }

  grad_kernel    <<<BATCH * NDIM, 256, 0, stream>>>(hds, mask, dep, Mmat, out_hp, rowent);
  finalize_kernel<<<BATCH,        512, 0, stream>>>(logdetw, rowent, out_hr, out_ent);
}